// MixMIL_42752104464903
// MI455X (gfx1250) — compile-verified
//
#include <hip/hip_runtime.h>
#include <hip/hip_bf16.h>
#include <math.h>

typedef __attribute__((ext_vector_type(2))) float v2f;
typedef __attribute__((ext_vector_type(8))) float v8f;

#define QDIM 128
#define SDIM 8
#define COLS 16   // [ u(8 cols) | z(8 cols) ]

// ---------------- kernel 0: b = rms(beta_z), Bmat = [beta_u | beta_z/b] ----------------
__global__ void __launch_bounds__(QDIM) k0_prep(const float* __restrict__ bu,
                                                const float* __restrict__ bz,
                                                float* __restrict__ bmat,
                                                float* __restrict__ bvec) {
    __shared__ float ssum[SDIM];
    __shared__ float sb[SDIM];
    int q = threadIdx.x;                      // 0..127
    if (q < SDIM) ssum[q] = 0.f;
    __syncthreads();
#pragma unroll
    for (int s = 0; s < SDIM; ++s) {
        float v = bz[q * SDIM + s];
        atomicAdd(&ssum[s], v * v);
    }
    __syncthreads();
    if (q < SDIM) {
        float b = sqrtf(ssum[q] / (float)QDIM);
        sb[q] = b;
        bvec[q] = b;
    }
    __syncthreads();
#pragma unroll
    for (int s = 0; s < SDIM; ++s) {
        bmat[q * COLS + s]        = bu[q * SDIM + s];
        bmat[q * COLS + SDIM + s] = bz[q * SDIM + s] / sb[s];
    }
}

// ---------------- kernel 1: fused GEMM  uz[I,16] = x[I,128] @ Bmat[128,16] via f32 WMMA ----------------
// one wave = one 16-row tile; 32 x V_WMMA_F32_16X16X4_F32 cover K=128
__global__ void __launch_bounds__(256) k1_gemm(const float* __restrict__ x,
                                               const float* __restrict__ bmat,
                                               float* __restrict__ uz,
                                               int I) {
    __shared__ float Bs[QDIM * COLS];         // 8 KB, loaded once per block
    int tid = threadIdx.x;
    for (int t = tid; t < QDIM * COLS; t += 256) Bs[t] = bmat[t];
    __syncthreads();

    int wave = tid >> 5;
    int lane = tid & 31;
    int hi   = lane >> 4;                     // 0: K pair {0,1}, 1: K pair {2,3}
    int lrow = lane & 15;

    long long row0 = ((long long)blockIdx.x * 8 + wave) * 16;
    if (row0 >= I) return;
    long long arow = row0 + lrow;
    if (arow >= I) arow = I - 1;              // clamp; stores handled below
    const float* ap = x + arow * QDIM + 2 * hi;

    // speculative prefetch of the next tile's rows (global_prefetch_b8)
    long long pfrow = (row0 + 16 < I) ? (row0 + 16) : row0;
    __builtin_prefetch(x + pfrow * QDIM + lane * 4, 0, 3);

    v8f acc = {};
#pragma unroll 8
    for (int kb = 0; kb < QDIM / 4; ++kb) {
        int k = kb * 4;
        // A 16x4 f32 fragment: lane<16 -> K=k,k+1 ; lane>=16 -> K=k+2,k+3 (contiguous b64 load)
        v2f a;
        a.x = ap[k + 0];
        a.y = ap[k + 1];
        // B 4x16 f32 fragment from LDS: VGPR0 -> K=k(+2*hi), VGPR1 -> K=k+1(+2*hi), N=lrow
        v2f b;
        b.x = Bs[(k + 2 * hi)     * COLS + lrow];
        b.y = Bs[(k + 2 * hi + 1) * COLS + lrow];
        acc = __builtin_amdgcn_wmma_f32_16x16x4_f32(
            /*neg_a=*/false, a, /*neg_b=*/false, b,
            /*c_mod=*/(short)0, acc, /*reuse_a=*/false, /*reuse_b=*/false);
    }
    // C/D layout: VGPR v, lanes 0-15 -> M=v, lanes 16-31 -> M=v+8, N=lane&15
    if (row0 + 16 <= I) {
        // full tile: single base pointer, 8 stores with immediate offsets (v*64 B)
        float* op = uz + (row0 + 8 * hi) * COLS + lrow;
#pragma unroll
        for (int v = 0; v < 8; ++v) op[v * COLS] = acc[v];
    } else {
        // remainder tile: guarded stores
#pragma unroll
        for (int v = 0; v < 8; ++v) {
            long long row = row0 + v + 8 * hi;
            if (row < I) uz[row * COLS + lrow] = acc[v];
        }
    }
}

// ---------------- kernel 2: bag start offsets via binary search (segments are sorted) ----------------
__global__ void k2_starts(const int* __restrict__ seg, int I,
                          const int* __restrict__ nbags, int* __restrict__ starts) {
    int N = *nbags;
    for (int n = blockIdx.x * blockDim.x + threadIdx.x; n <= N;
         n += gridDim.x * blockDim.x) {
        int lo = 0, hi = I;                   // lower_bound(seg, n)
        while (lo < hi) {
            int mid = (lo + hi) >> 1;
            if (seg[mid] < n) lo = mid + 1; else hi = mid;
        }
        starts[n] = lo;
    }
}

// ---------------- kernel 3: per-bag softmax-weighted pooling (one wave per bag) ----------------
// 8 independent waves per block; lane -> (r = lane/8 row-slice, s = lane%8 column);
// cross-r reduce via shfl_xor(8,16) -- all reductions intra-wave, no block sync.
__global__ void __launch_bounds__(256) k3_pool(const float* __restrict__ uz,
                                               const int* __restrict__ starts,
                                               const int* __restrict__ nbags,
                                               float* __restrict__ xmil) {
    int N = *nbags;
    int lane = threadIdx.x & 31;
    int wav  = threadIdx.x >> 5;
    int r = lane >> 3, s = lane & 7;
    for (int n = blockIdx.x * 8 + wav; n < N; n += gridDim.x * 8) {
        int st = starts[n], en = starts[n + 1];
        float m = -__builtin_inff();
        for (int idx = st + r; idx < en; idx += 4)
            m = fmaxf(m, uz[(long long)idx * COLS + s]);
        m = fmaxf(m, __shfl_xor(m, 8, 32));
        m = fmaxf(m, __shfl_xor(m, 16, 32));
        float se = 0.f, sez = 0.f;
        for (int idx = st + r; idx < en; idx += 4) {
            float u = uz[(long long)idx * COLS + s];
            float z = uz[(long long)idx * COLS + SDIM + s];
            float e = __expf(u - m);
            se += e;
            sez += e * z;
        }
        se  += __shfl_xor(se, 8, 32);   sez += __shfl_xor(sez, 8, 32);
        se  += __shfl_xor(se, 16, 32);  sez += __shfl_xor(sez, 16, 32);
        if (lane < SDIM)
            xmil[(long long)n * SDIM + s] = (en > st) ? (sez / se) : 0.f;
    }
}

// ---------------- kernel 4: standardize across bags (ddof=1), rescale by b ----------------
__global__ void __launch_bounds__(256) k4_norm(const float* __restrict__ xmil,
                                               const float* __restrict__ bvec,
                                               const int* __restrict__ nbags,
                                               float* __restrict__ out) {
    __shared__ float s_sum[256];
    __shared__ float s_sq[256];
    int N = *nbags;
    int s = blockIdx.x;                       // one block per column
    float sum = 0.f, sq = 0.f;
    for (int n = threadIdx.x; n < N; n += 256) {
        float v = xmil[(long long)n * SDIM + s];
        sum += v; sq += v * v;
    }
    s_sum[threadIdx.x] = sum;
    s_sq[threadIdx.x]  = sq;
    __syncthreads();
    for (int off = 128; off > 0; off >>= 1) {
        if (threadIdx.x < off) {
            s_sum[threadIdx.x] += s_sum[threadIdx.x + off];
            s_sq[threadIdx.x]  += s_sq[threadIdx.x + off];
        }
        __syncthreads();
    }
    float mean = s_sum[0] / (float)N;
    float var  = (s_sq[0] - (float)N * mean * mean) / (float)(N - 1);
    float inv  = bvec[s] * rsqrtf(var);
    for (int n = threadIdx.x; n < N; n += 256)
        out[(long long)n * SDIM + s] = (xmil[(long long)n * SDIM + s] - mean) * inv;
}

extern "C" void kernel_launch(void* const* d_in, const int* in_sizes, int n_in,
                              void* d_out, int out_size, void* d_ws, size_t ws_size,
                              hipStream_t stream) {
    const float* x     = (const float*)d_in[0];   // [I,128]
    const float* bu    = (const float*)d_in[1];   // [128,1,8]
    const float* bz    = (const float*)d_in[2];   // [128,1,8]
    const int*   seg   = (const int*)d_in[3];     // [I] sorted bag ids
    const int*   nbags = (const int*)d_in[4];     // scalar N (device)
    float* out = (float*)d_out;                   // [N,1,8]

    int I = in_sizes[3];

    // workspace layout
    char* ws = (char*)d_ws;
    float* bmat   = (float*)(ws);                 // 128*16 f32      (8 KB)
    float* bvec   = (float*)(ws + 8192);          // 8 f32
    int*   starts = (int*)  (ws + (64 << 10));    // (N+1) int
    float* xmil   = (float*)(ws + (2 << 20));     // N*8 f32
    float* uz     = (float*)(ws + (4 << 20));     // I*16 f32        (32.8 MB)

    k0_prep<<<1, QDIM, 0, stream>>>(bu, bz, bmat, bvec);

    int tiles   = (I + 15) / 16;
    int blocks1 = (tiles + 7) / 8;                // 8 waves (tiles) per block
    k1_gemm<<<blocks1, 256, 0, stream>>>(x, bmat, uz, I);

    k2_starts<<<256, 256, 0, stream>>>(seg, I, nbags, starts);
    k3_pool<<<512, 256, 0, stream>>>(uz, starts, nbags, xmil);
    k4_norm<<<SDIM, 256, 0, stream>>>(xmil, bvec, nbags, out);
}